// FilterModule_83940840833425
// MI455X (gfx1250) — compile-verified
//
#include <hip/hip_runtime.h>
#include <math.h>

// LTV FIR + overlap-add, restructured as per-group dense GEMMs on the fp32 WMMA pipe.
//   out[f,t] = w[t] * sum_s (w[s]*src[40f+s-80]) * ((1-frac_f)*C[256+s-t, lo_f] + frac_f*C[256+s-t, lo_f+1])
// Frames with the same lo (= floor(f*19999/1599999), ~80 frames) share the two 80x80
// Toeplitz matrices -> batched GEMM (M<=96, K=80, N=2x80) per group, V_WMMA_F32_16X16X4_F32.

#define FS       80
#define HOPSZ    40
#define TCOLS    20000
#define NFRAMES  40002
#define NGROUPS  500
#define SRCLEN   1600000
#define OUTLEN   1600000
#define LDS_STRIDE 84     // 336 B row stride: 8B aligned, conflict-free across 16 rows
#define MAX_ROWS   96     // 6 M-tiles of 16 covers max group size (81)
#define NTHREADS   320    // 10 wave32 waves

typedef __attribute__((ext_vector_type(2))) float v2f;
typedef __attribute__((ext_vector_type(8))) float v8f;

__device__ __forceinline__ int lo_of(int f) {
  const float R = (float)(19999.0 / 1599999.0);   // matches reference fp32 math
  return (int)floorf((float)f * R);
}

__device__ __forceinline__ float hann(int n) {
  return 0.5f * (1.0f - __cosf(0.07853981633974483f * (float)n)); // 2*pi/80
}

template<bool USE_WS>
__global__ __launch_bounds__(NTHREADS)
void ltv_group_kernel(const float* __restrict__ src,
                      const float* __restrict__ fc,
                      float* __restrict__ ws,
                      float* __restrict__ out) {
  __shared__ float sF[MAX_ROWS * LDS_STRIDE]; // windowed frames of this group
  __shared__ float sClo[160];                 // filter column lo, taps 177..335
  __shared__ float sChi[160];                 // filter column lo+1

  const int j   = blockIdx.x;                 // lo value handled by this workgroup
  const int tid = threadIdx.x;
  const float R = (float)(19999.0 / 1599999.0);

  // ---- frame range [f0, f1] with fp32 floor(f*R) == j (bit-consistent w/ reference) ----
  int guess = (int)(((long long)j * 1599999LL) / 19999LL);
  int f0 = guess;
  for (int f = (guess > 4 ? guess - 4 : 0); f <= guess + 4; ++f)
    if (lo_of(f) == j) { f0 = f; break; }
  int f1;
  if (j == NGROUPS - 1) {
    f1 = NFRAMES - 1;
  } else {
    int g2 = (int)(((long long)(j + 1) * 1599999LL) / 19999LL);
    f1 = g2 + 4;
    for (int f = (g2 > 4 ? g2 - 4 : 0); f <= g2 + 4; ++f)
      if (lo_of(f) == j + 1) { f1 = f - 1; break; }
  }
  const int count  = f1 - f0 + 1;             // 80 or 81
  const int nTiles = (count + 15) >> 4;       // 5 or 6 M-tiles
  const int rows   = nTiles * 16;

  // ---- stage the two live filter columns (only taps 177..335 can touch the frame) ----
  for (int i = tid; i < 160; i += NTHREADS) {
    float lo = 0.f, hi = 0.f;
    if (i < 159) {
      int k = 177 + i;
      lo = fc[k * TCOLS + j];                 // filter_coeffs[0, k, j]
      hi = fc[k * TCOLS + j + 1];             // hi = lo+1 always (lo <= 499 << 19999)
    }
    sClo[i] = lo;
    sChi[i] = hi;
  }
  // ---- stage windowed frames (zero-pad partial last tile and signal edges) ----
  for (int idx = tid; idx < rows * FS; idx += NTHREADS) {
    int r = idx / FS;
    int s = idx - r * FS;
    int f = f0 + r;
    float v = 0.f;
    if (f <= f1) {
      int g = f * HOPSZ + s - FS;             // unpadded source index
      if (g >= 0 && g < SRCLEN) v = src[g] * hann(s);
    }
    sF[r * LDS_STRIDE + s] = v;
  }
  __syncthreads();

  // ---- WMMA fragment addressing (16x16x4 f32) ----
  // A (16x4 f32): M = lane&15 for all lanes; v0/v1 hold K = {0,1} (lanes 0-15) / {2,3} (16-31)
  // B (4x16 f32): N = lane&15; v0/v1 hold K rows {0,1} / {2,3}.  B[s,t] = C[79 + s - t] in LDS
  // C/D (16x16 f32): VGPR r, lanes 0-15 -> M=r, lanes 16-31 -> M=r+8; N = lane&15.
  const int wv    = tid >> 5;
  const int lane  = tid & 31;
  const int l15   = lane & 15;
  const int n     = wv % 5;                   // output-time tile: t in [16n, 16n+16)
  const int t     = 16 * n + l15;
  const int khalf = (lane < 16) ? 0 : 2;
  const float wt  = hann(t);
  const float* arow_base = &sF[l15 * LDS_STRIDE + khalf];
  const float* blo = &sClo[79 - t + khalf];   // 79 - t >= 0
  const float* bhi = &sChi[79 - t + khalf];

  for (int m = wv / 5; m < nTiles; m += 2) {  // waves 0-4: m=0,2,4; waves 5-9: m=1,3,5
    const float* arow = arow_base + m * (16 * LDS_STRIDE);
    v8f acc_lo = {0.f, 0.f, 0.f, 0.f, 0.f, 0.f, 0.f, 0.f};
    v8f acc_hi = acc_lo;
    #pragma unroll
    for (int k = 0; k < 20; ++k) {
      v2f a, bl, bh;
      a.x  = arow[4 * k];  a.y  = arow[4 * k + 1];
      bl.x = blo[4 * k];   bl.y = blo[4 * k + 1];
      bh.x = bhi[4 * k];   bh.y = bhi[4 * k + 1];
      acc_lo = __builtin_amdgcn_wmma_f32_16x16x4_f32(false, a, false, bl,
                                                     (short)0, acc_lo, false, false);
      acc_hi = __builtin_amdgcn_wmma_f32_16x16x4_f32(false, a, false, bh,
                                                     (short)0, acc_hi, false, false);
    }
    // ---- interpolate lo/hi results per frame, apply synthesis window, write out ----
    const int mbase = (lane < 16) ? 0 : 8;
    #pragma unroll
    for (int r8 = 0; r8 < 8; ++r8) {
      int f = f0 + 16 * m + mbase + r8;
      if (f <= f1) {
        float p    = (float)f * R;            // fp32, as in reference
        float frac = p - (float)j;
        float olo  = acc_lo[r8];
        float val  = wt * (olo + (acc_hi[r8] - olo) * frac);
        if (USE_WS) {
          ws[f * FS + t] = val;               // per-frame output, gathered later
        } else {
          int oi = f * HOPSZ + t - FS;        // crop [fs:-fs]
          if (oi >= 0 && oi < OUTLEN) atomicAdd(&out[oi], val);
        }
      }
    }
  }
}

// Deterministic overlap-add: padded position q = i+80 receives exactly frames
// f = q/40 (tap t = q%40) and f-1 (tap t = q%40 + 40); both always valid.
__global__ __launch_bounds__(256)
void overlap_add_gather(const float* __restrict__ ws, float* __restrict__ out) {
  int i = blockIdx.x * blockDim.x + threadIdx.x;
  if (i >= OUTLEN) return;
  int q  = i + FS;
  int f  = q / HOPSZ;                         // in [2, 40001]
  int t  = q - f * HOPSZ;                     // in [0, 40)
  out[i] = ws[f * FS + t] + ws[(f - 1) * FS + t + HOPSZ];
}

__global__ __launch_bounds__(256)
void zero_out_kernel(float* __restrict__ out) {
  int i = blockIdx.x * blockDim.x + threadIdx.x;
  if (i < OUTLEN) out[i] = 0.f;
}

extern "C" void kernel_launch(void* const* d_in, const int* in_sizes, int n_in,
                              void* d_out, int out_size, void* d_ws, size_t ws_size,
                              hipStream_t stream) {
  const float* src = (const float*)d_in[0];   // source: 1,600,000 fp32
  const float* fc  = (const float*)d_in[1];   // filter_coeffs: 513 x 20000 fp32
  float* out = (float*)d_out;

  const size_t ws_need = (size_t)NFRAMES * FS * sizeof(float); // 12.8 MB
  if (ws_size >= ws_need) {
    float* wsf = (float*)d_ws;
    ltv_group_kernel<true><<<NGROUPS, NTHREADS, 0, stream>>>(src, fc, wsf, out);
    overlap_add_gather<<<(OUTLEN + 255) / 256, 256, 0, stream>>>(wsf, out);
  } else {
    // fallback: accumulate directly into zeroed output with f32 atomics
    zero_out_kernel<<<(OUTLEN + 255) / 256, 256, 0, stream>>>(out);
    ltv_group_kernel<false><<<NGROUPS, NTHREADS, 0, stream>>>(src, fc, nullptr, out);
  }
}